// GNN_Match_XGB_21053929685025
// MI455X (gfx1250) — compile-verified
//
#include <hip/hip_runtime.h>

typedef __attribute__((ext_vector_type(2))) float v2f;
typedef __attribute__((ext_vector_type(8))) float v8f;

#define N_NODES 65536
#define N_EDGES 1048576
#define DIM     64

// Order-preserving float -> int key (monotone under signed int compare).
__device__ __forceinline__ int   f2key(float f) {
  int i = __float_as_int(f);
  return i >= 0 ? i : (i ^ 0x7FFFFFFF);
}
__device__ __forceinline__ float key2f(int i) {
  return __int_as_float(i >= 0 ? i : (i ^ 0x7FFFFFFF));
}

// ---------------------------------------------------------------------------
// K0: zero h_out (d_out is poisoned each run), init segment-max keys + denom.
// ---------------------------------------------------------------------------
__global__ void k0_init(float* __restrict__ out, int* __restrict__ emax_i,
                        float* __restrict__ denom) {
  int i = blockIdx.x * blockDim.x + threadIdx.x;   // 0 .. N_NODES*DIM-1
  out[i] = 0.0f;
  if (i < N_NODES) {
    emax_i[i] = (int)0x80000000;   // below every real float key
    denom[i]  = 0.0f;
  }
}

// ---------------------------------------------------------------------------
// K1: z = h @ fc_w^T via V_WMMA_F32_16X16X4_F32.
// One wave computes one 16x16 tile of z; K=64 -> 16 chained WMMAs.
// A tile = rows of h; B tile (4x16 of fc_w^T) = rows of fc_w (same fetch form).
// ---------------------------------------------------------------------------
__global__ void k1_gemm(const float* __restrict__ h, const float* __restrict__ w,
                        float* __restrict__ z) {
  int wave = (blockIdx.x * blockDim.x + threadIdx.x) >> 5;  // 0 .. 16383
  int lane = threadIdx.x & 31;
  int mt = wave >> 2;                 // 4096 M-tiles of 16 rows
  int nt = wave & 3;                  // 4 N-tiles of 16 cols
  int r16   = lane & 15;
  int khalf = (lane >> 4) << 1;       // 0 or 2

  const float* arow = h + (size_t)(mt * 16 + r16) * DIM;  // h row (A, MxK)
  const float* brow = w + (size_t)(nt * 16 + r16) * DIM;  // fc_w row n (B=W^T, KxN)

  v8f c = {};
#pragma unroll
  for (int s = 0; s < 16; ++s) {
    int k = (s << 2) + khalf;         // 8-byte aligned (k even)
    v2f a = *(const v2f*)(arow + k);
    v2f b = *(const v2f*)(brow + k);
    c = __builtin_amdgcn_wmma_f32_16x16x4_f32(
            /*neg_a=*/false, a, /*neg_b=*/false, b,
            /*c_mod=*/(short)0, c, /*reuse_a=*/false, /*reuse_b=*/false);
  }

  // C layout: VGPR r -> M = r + 8*(lane>=16), N = lane%16.
  int mbase = mt * 16 + ((lane >> 4) << 3);
  int n     = nt * 16 + r16;
#pragma unroll
  for (int r = 0; r < 8; ++r)
    z[(size_t)(mbase + r) * DIM + n] = c[r];
}

// ---------------------------------------------------------------------------
// K2: per-node attention scores es = z.a_src, ed = z.a_dst (z streamed once).
// ---------------------------------------------------------------------------
__global__ void k2_scores(const float* __restrict__ z, const float* __restrict__ attn,
                          float* __restrict__ es, float* __restrict__ ed) {
  int n = blockIdx.x * blockDim.x + threadIdx.x;   // 0 .. N_NODES-1
  const float4* zr = (const float4*)(z + (size_t)n * DIM);
  const float4* as = (const float4*)(attn);        // attn_w[0, 0:64]
  const float4* ad = (const float4*)(attn + DIM);  // attn_w[0, 64:128]
  float s = 0.0f, d = 0.0f;
#pragma unroll
  for (int k = 0; k < DIM / 4; ++k) {
    float4 zv = zr[k], av = as[k], dv = ad[k];
    s += zv.x * av.x + zv.y * av.y + zv.z * av.z + zv.w * av.w;
    d += zv.x * dv.x + zv.y * dv.y + zv.z * dv.z + zv.w * dv.w;
  }
  es[n] = s;
  ed[n] = d;
}

// ---------------------------------------------------------------------------
// K3: per-edge e = leaky_relu(es[src]+ed[dst]); segment max into emax_i[dst].
// ---------------------------------------------------------------------------
__global__ void k3_edge_max(const float* __restrict__ es, const float* __restrict__ ed,
                            const int* __restrict__ src, const int* __restrict__ dst,
                            float* __restrict__ e_buf, int* __restrict__ emax_i) {
  int t = blockIdx.x * blockDim.x + threadIdx.x;   // 0 .. N_EDGES-1
  int d = dst[t];
  float e = es[src[t]] + ed[d];
  e = e >= 0.0f ? e : 0.01f * e;
  e_buf[t] = e;
  atomicMax(&emax_i[d], f2key(e));
}

// ---------------------------------------------------------------------------
// K4: per-edge w = exp(e - emax[dst]); segment sum into denom[dst].
// ---------------------------------------------------------------------------
__global__ void k4_edge_exp(const float* __restrict__ e_buf, const int* __restrict__ dst,
                            const int* __restrict__ emax_i, float* __restrict__ w_buf,
                            float* __restrict__ denom) {
  int t = blockIdx.x * blockDim.x + threadIdx.x;
  int d = dst[t];
  float m = key2f(emax_i[d]);
  if (!isfinite(m)) m = 0.0f;
  float wv = __expf(e_buf[t] - m);
  w_buf[t] = wv;
  atomicAdd(&denom[d], wv);
}

// ---------------------------------------------------------------------------
// K5: h_out[dst] += (w/denom[dst]) * z[src]. 16 lanes per edge, float4 each:
// one edge's 256B gather is a coalesced half-wave load; adds hit L2 atomics.
// ---------------------------------------------------------------------------
__global__ void k5_scatter(const float* __restrict__ z, const float* __restrict__ w_buf,
                           const float* __restrict__ denom, const int* __restrict__ src,
                           const int* __restrict__ dst, float* __restrict__ out) {
  int t    = blockIdx.x * blockDim.x + threadIdx.x; // 0 .. N_EDGES*16-1
  int edge = t >> 4;
  int part = t & 15;                                // 4 floats per part
  int s = src[edge], d = dst[edge];
  float dn = denom[d];
  float alpha = w_buf[edge] / (dn > 0.0f ? dn : 1.0f);
  float4 zv = *(const float4*)(z + (size_t)s * DIM + part * 4);
  float* o = out + (size_t)d * DIM + part * 4;
  atomicAdd(o + 0, alpha * zv.x);
  atomicAdd(o + 1, alpha * zv.y);
  atomicAdd(o + 2, alpha * zv.z);
  atomicAdd(o + 3, alpha * zv.w);
}

// ---------------------------------------------------------------------------
extern "C" void kernel_launch(void* const* d_in, const int* in_sizes, int n_in,
                              void* d_out, int out_size, void* d_ws, size_t ws_size,
                              hipStream_t stream) {
  const float* h     = (const float*)d_in[0];   // (65536, 64)
  const float* fc_w  = (const float*)d_in[1];   // (64, 64)
  const float* attn  = (const float*)d_in[2];   // (1, 128)
  const int*   src   = (const int*)d_in[3];     // (1048576,)
  const int*   dst   = (const int*)d_in[4];     // (1048576,)
  float*       out   = (float*)d_out;           // (65536, 64)

  // Workspace carve-up (256B aligned regions), 25 MB total.
  char* ws = (char*)d_ws;
  size_t off = 0;
  float* z      = (float*)(ws + off); off += (size_t)N_NODES * DIM * 4;   // 16 MB
  float* es     = (float*)(ws + off); off += (size_t)N_NODES * 4;
  float* ed     = (float*)(ws + off); off += (size_t)N_NODES * 4;
  int*   emax_i = (int*)  (ws + off); off += (size_t)N_NODES * 4;
  float* denom  = (float*)(ws + off); off += (size_t)N_NODES * 4;
  float* e_buf  = (float*)(ws + off); off += (size_t)N_EDGES * 4;         // 4 MB
  float* w_buf  = (float*)(ws + off); off += (size_t)N_EDGES * 4;         // 4 MB
  (void)off; (void)ws_size; (void)in_sizes; (void)n_in; (void)out_size;

  const int B = 256;
  k0_init   <<<(N_NODES * DIM) / B, B, 0, stream>>>(out, emax_i, denom);
  // 16384 waves (4096 M-tiles x 4 N-tiles), 8 waves/block.
  k1_gemm   <<<(N_NODES / 16) * 4 * 32 / B, B, 0, stream>>>(h, fc_w, z);
  k2_scores <<<N_NODES / B, B, 0, stream>>>(z, attn, es, ed);
  k3_edge_max<<<N_EDGES / B, B, 0, stream>>>(es, ed, src, dst, e_buf, emax_i);
  k4_edge_exp<<<N_EDGES / B, B, 0, stream>>>(e_buf, dst, emax_i, w_buf, denom);
  k5_scatter<<<(size_t)N_EDGES * 16 / B, B, 0, stream>>>(z, w_buf, denom, src, dst, out);
}